// GraphNet_25288767439626
// MI455X (gfx1250) — compile-verified
//
#include <hip/hip_runtime.h>

#define N_NODES 50000
#define N_EDGES 800000
#define NODE_IN 16
#define EDGE_IN 3
#define G_DIM   8
#define LATENT  128
#define KPAD    28   // 25 real K (16 node + 3 sent + 3 recv + cnt_s + cnt_r + 1), padded to 7x4

typedef __attribute__((ext_vector_type(2))) float v2f;
typedef __attribute__((ext_vector_type(8))) float v8f;

// Workspace layout (float offsets)
#define WS_MF    0                    // fused layer-1 matrix M: [KPAD][LATENT]
#define WS_D     (KPAD * LATENT)      // fused W2@W_dn: [LATENT]
#define WS_D0    (WS_D + LATENT)      // scalar bias for nodes_out
#define WS_WE    (WS_D0 + 1)          // fused W_ee@W_de: [3]
#define WS_WE0   (WS_WE + 3)          // scalar bias for edges_out
#define WS_SENT  3720                 // [N_NODES][4]: sum(e0,e1,e2), count
#define WS_RECV  (WS_SENT + 4 * N_NODES)

__global__ void zero_acc_kernel(float* __restrict__ ws) {
    int i = blockIdx.x * blockDim.x + threadIdx.x;
    if (i < 8 * N_NODES) ws[WS_SENT + i] = 0.0f;
}

// One block of 128 threads: thread t builds column t of every fused weight.
__global__ void fuse_weights_kernel(const float* __restrict__ W_en, const float* __restrict__ b_en,
                                    const float* __restrict__ W_ee, const float* __restrict__ b_ee,
                                    const float* __restrict__ W1,   const float* __restrict__ b1,
                                    const float* __restrict__ W2,   const float* __restrict__ b2,
                                    const float* __restrict__ W_dn, const float* __restrict__ b_dn,
                                    const float* __restrict__ W_de, const float* __restrict__ b_de,
                                    const float* __restrict__ g,
                                    float* __restrict__ ws) {
    const int t = threadIdx.x;           // 0..127 (column of LATENT)
    // rows 0..15: W_en @ W1[0:128]
    for (int i = 0; i < NODE_IN; ++i) {
        float s = 0.f;
        for (int j = 0; j < LATENT; ++j) s += W_en[i * LATENT + j] * W1[j * LATENT + t];
        ws[WS_MF + i * LATENT + t] = s;
    }
    // rows 16..18: W_ee @ W1[128:256]   rows 19..21: W_ee @ W1[256:384]
    for (int i = 0; i < EDGE_IN; ++i) {
        float sb = 0.f, sc = 0.f;
        for (int j = 0; j < LATENT; ++j) {
            float w = W_ee[i * LATENT + j];
            sb += w * W1[(LATENT     + j) * LATENT + t];
            sc += w * W1[(2 * LATENT + j) * LATENT + t];
        }
        ws[WS_MF + (16 + i) * LATENT + t] = sb;
        ws[WS_MF + (19 + i) * LATENT + t] = sc;
    }
    // row 22: b_ee @ W1b (scaled by sender count)  row 23: b_ee @ W1c (receiver count)
    {
        float su = 0.f, sv = 0.f;
        for (int j = 0; j < LATENT; ++j) {
            float bj = b_ee[j];
            su += bj * W1[(LATENT     + j) * LATENT + t];
            sv += bj * W1[(2 * LATENT + j) * LATENT + t];
        }
        ws[WS_MF + 22 * LATENT + t] = su;
        ws[WS_MF + 23 * LATENT + t] = sv;
    }
    // row 24: b_en @ W1a + globals @ W1d + b1   (multiplied by constant z=1)
    {
        float s = b1[t];
        for (int j = 0; j < LATENT; ++j) s += b_en[j] * W1[j * LATENT + t];
        for (int j = 0; j < G_DIM;  ++j) s += g[j] * W1[(3 * LATENT + j) * LATENT + t];
        ws[WS_MF + 24 * LATENT + t] = s;
    }
    // rows 25..27: K padding
    ws[WS_MF + 25 * LATENT + t] = 0.f;
    ws[WS_MF + 26 * LATENT + t] = 0.f;
    ws[WS_MF + 27 * LATENT + t] = 0.f;
    // D = W2 @ W_dn
    {
        float s = 0.f;
        for (int j = 0; j < LATENT; ++j) s += W2[t * LATENT + j] * W_dn[j];
        ws[WS_D + t] = s;
    }
    if (t == 0) {   // d0 = b2 . W_dn + b_dn
        float s = b_dn[0];
        for (int j = 0; j < LATENT; ++j) s += b2[j] * W_dn[j];
        ws[WS_D0] = s;
    }
    if (t < 3) {    // We = W_ee @ W_de
        float s = 0.f;
        for (int j = 0; j < LATENT; ++j) s += W_ee[t * LATENT + j] * W_de[j];
        ws[WS_WE + t] = s;
    }
    if (t == 3) {   // we0 = b_ee . W_de + b_de
        float s = b_de[0];
        for (int j = 0; j < LATENT; ++j) s += b_ee[j] * W_de[j];
        ws[WS_WE0] = s;
    }
}

// Per-edge: scatter 3 floats + count into sender/receiver accumulators (L2-resident,
// 1.6 MB total) and emit the fused [3 -> 1] edge decoder output.
__global__ void edge_kernel(const float* __restrict__ edges, const int* __restrict__ senders,
                            const int* __restrict__ receivers, float* __restrict__ ws,
                            float* __restrict__ edges_out) {
    int i = blockIdx.x * blockDim.x + threadIdx.x;
    if (i >= N_EDGES) return;
    float e0 = edges[3 * i + 0];
    float e1 = edges[3 * i + 1];
    float e2 = edges[3 * i + 2];
    int s = senders[i];
    int r = receivers[i];
    float* sp = ws + WS_SENT + 4 * s;
    atomicAdd(sp + 0, e0); atomicAdd(sp + 1, e1); atomicAdd(sp + 2, e2); atomicAdd(sp + 3, 1.0f);
    float* rp = ws + WS_RECV + 4 * r;
    atomicAdd(rp + 0, e0); atomicAdd(rp + 1, e1); atomicAdd(rp + 2, e2); atomicAdd(rp + 3, 1.0f);
    edges_out[i] = e0 * ws[WS_WE + 0] + e1 * ws[WS_WE + 1] + e2 * ws[WS_WE + 2] + ws[WS_WE0];
}

// One wave (32 lanes) per 16-node tile. Dense GEMM z[16,28] @ M[28,128] via
// V_WMMA_F32_16X16X4_F32 (8 N-tiles x 7 K-steps = 56 WMMAs), then fused
// relu + dot(D) reduction -> one scalar per node.
__global__ void __launch_bounds__(32)
node_kernel(const float* __restrict__ nodes, const float* __restrict__ ws,
            float* __restrict__ nodes_out) {
    __shared__ float zl[16][KPAD];
    __shared__ float red[16][16];
    const int lane = threadIdx.x;
    const int m0   = blockIdx.x * 16;
    const int half = lane >> 4;      // 0: lanes 0-15, 1: lanes 16-31
    const int m    = lane & 15;

    // Stage augmented node-input tile z into LDS.
    for (int i = lane; i < 16 * KPAD; i += 32) {
        int r = i / KPAD, c = i % KPAD;
        int node = m0 + r;
        float v = 0.f;
        if (node < N_NODES) {
            if      (c < 16) v = nodes[node * NODE_IN + c];
            else if (c < 19) v = ws[WS_SENT + node * 4 + (c - 16)];
            else if (c < 22) v = ws[WS_RECV + node * 4 + (c - 19)];
            else if (c == 22) v = ws[WS_SENT + node * 4 + 3];  // sender count
            else if (c == 23) v = ws[WS_RECV + node * 4 + 3];  // receiver count
            else if (c == 24) v = 1.0f;                        // constant -> bias row
        }
        zl[r][c] = v;
    }
    __syncthreads();

    // A fragments: 16x4 f32 layout. VGPR0 = K+0 (lanes 0-15) / K+2 (lanes 16-31),
    // VGPR1 = K+1 / K+3; M = lane&15.
    v2f afrag[7];
#pragma unroll
    for (int k = 0; k < 7; ++k) {
        afrag[k].x = zl[m][4 * k + 2 * half + 0];
        afrag[k].y = zl[m][4 * k + 2 * half + 1];
    }

    const float* Mf = ws + WS_MF;
    float part[8];
#pragma unroll
    for (int j = 0; j < 8; ++j) part[j] = 0.f;

#pragma unroll
    for (int nt = 0; nt < 8; ++nt) {
        v8f acc = {};
#pragma unroll
        for (int k = 0; k < 7; ++k) {
            v2f b;   // B 4x16: symmetric layout, rows K+0/K+2 then K+1/K+3 by lane half
            b.x = Mf[(4 * k + 2 * half + 0) * LATENT + nt * 16 + m];
            b.y = Mf[(4 * k + 2 * half + 1) * LATENT + nt * 16 + m];
            acc = __builtin_amdgcn_wmma_f32_16x16x4_f32(
                false, afrag[k], false, b, (short)0, acc, false, false);
        }
        float dcol = ws[WS_D + nt * 16 + m];
#pragma unroll
        for (int j = 0; j < 8; ++j) {
            float x = acc[j];
            x = x > 0.f ? x : 0.f;           // relu
            part[j] += x * dcol;             // fused (W2 @ W_dn) dot
        }
    }

    // C/D layout: VGPR j holds row j (lanes 0-15) / row j+8 (lanes 16-31), col = m.
#pragma unroll
    for (int j = 0; j < 8; ++j) red[half * 8 + j][m] = part[j];
    __syncthreads();

    if (lane < 16) {
        int node = m0 + lane;
        if (node < N_NODES) {
            float s = ws[WS_D0];
#pragma unroll
            for (int c = 0; c < 16; ++c) s += red[lane][c];
            nodes_out[node] = s;
        }
    }
}

extern "C" void kernel_launch(void* const* d_in, const int* in_sizes, int n_in,
                              void* d_out, int out_size, void* d_ws, size_t ws_size,
                              hipStream_t stream) {
    const float* nodes     = (const float*)d_in[0];
    const float* edges     = (const float*)d_in[1];
    const float* globals_  = (const float*)d_in[2];
    const int*   senders   = (const int*)d_in[3];
    const int*   receivers = (const int*)d_in[4];
    const float* W_en = (const float*)d_in[5];
    const float* b_en = (const float*)d_in[6];
    const float* W_ee = (const float*)d_in[7];
    const float* b_ee = (const float*)d_in[8];
    const float* W1   = (const float*)d_in[9];
    const float* b1   = (const float*)d_in[10];
    const float* W2   = (const float*)d_in[11];
    const float* b2   = (const float*)d_in[12];
    const float* W_dn = (const float*)d_in[13];
    const float* b_dn = (const float*)d_in[14];
    const float* W_de = (const float*)d_in[15];
    const float* b_de = (const float*)d_in[16];

    float* out       = (float*)d_out;
    float* ws        = (float*)d_ws;
    float* nodes_out = out;               // [N_NODES]
    float* edges_out = out + N_NODES;     // [N_EDGES]

    zero_acc_kernel<<<(8 * N_NODES + 255) / 256, 256, 0, stream>>>(ws);
    fuse_weights_kernel<<<1, 128, 0, stream>>>(W_en, b_en, W_ee, b_ee, W1, b1,
                                               W2, b2, W_dn, b_dn, W_de, b_de,
                                               globals_, ws);
    edge_kernel<<<(N_EDGES + 255) / 256, 256, 0, stream>>>(edges, senders, receivers,
                                                           ws, edges_out);
    node_kernel<<<(N_NODES + 15) / 16, 32, 0, stream>>>(nodes, ws, nodes_out);
}